// RetinaNet_40544491274414
// MI455X (gfx1250) — compile-verified
//
#include <hip/hip_runtime.h>
#include <stdint.h>

// ---------------- problem constants (from the reference) ----------------
#define NCLS    80
#define ATOT    120087
#define NLVL    5
#define NIMG    8
#define NCOMBO  (NIMG * NLVL)       // 40 (image, level) pairs
#define TOPK    1000
#define NDET    300
#define NCAND   (NLVL * TOPK)       // 5000 candidates per image
#define CAP     8192                // candidate buffer capacity per combo
#define NBIN    2048                // coarse histogram bins (top 11 key bits)
#define IMGSZ   800.0f
#define TH_LOGIT  (-2.94443897916644f)   // log(0.05/0.95): sigmoid(x)>0.05  <=>  x>TH
#define BOXCLAMP  (4.13516655674235601f) // log(1000/16)
#define NMS_TH    0.5f

typedef __attribute__((ext_vector_type(2))) float        v2f;
typedef __attribute__((ext_vector_type(8))) float        v8f;
typedef __attribute__((ext_vector_type(4))) unsigned int v4u;
typedef __attribute__((ext_vector_type(4))) int          v4i;
typedef __attribute__((ext_vector_type(8))) int          v8i;

static __device__ const int d_lvl_size[NLVL] = {90000, 22500, 5625, 1521, 441};
static __device__ const int d_lvl_off [NLVL] = {0, 90000, 112500, 118125, 119646};

// Monotone map: larger float  <=>  larger uint32 key
__device__ __forceinline__ uint32_t fkey(float x) {
  uint32_t u = __float_as_uint(x);
  return (u & 0x80000000u) ? ~u : (u | 0x80000000u);
}
__device__ __forceinline__ float key2f(uint32_t k) {
  uint32_t u = (k & 0x80000000u) ? (k ^ 0x80000000u) : ~k;
  return __uint_as_float(u);
}

// ---------------- CDNA5 async / tensor data paths ----------------
__device__ __forceinline__ void async_g2l_b128(uint32_t lds_addr, const void* gaddr) {
  asm volatile("global_load_async_to_lds_b128 %0, %1, off"
               :: "v"(lds_addr), "v"(gaddr) : "memory");
}
__device__ __forceinline__ void async_g2l_b32(uint32_t lds_addr, const void* gaddr) {
  asm volatile("global_load_async_to_lds_b32 %0, %1, off"
               :: "v"(lds_addr), "v"(gaddr) : "memory");
}
__device__ __forceinline__ void wait_async0() {
#if __has_builtin(__builtin_amdgcn_s_wait_asynccnt)
  __builtin_amdgcn_s_wait_asynccnt(0);
#else
  asm volatile("s_wait_asynccnt 0" ::: "memory");
#endif
}
__device__ __forceinline__ void wait_tensor0() {
#if __has_builtin(__builtin_amdgcn_s_wait_tensorcnt)
  __builtin_amdgcn_s_wait_tensorcnt(0);
#else
  asm volatile("s_wait_tensorcnt 0" ::: "memory");
#endif
}

// ---------------- kernel 0: zero histograms + meta ----------------
__global__ void k_zero(uint32_t* __restrict__ p, int n) {
  int i = blockIdx.x * blockDim.x + threadIdx.x;
  int stride = gridDim.x * blockDim.x;
  for (; i < n; i += stride) p[i] = 0u;
}

// ---------------- kernel 1: per-combo coarse histogram of logit keys ----------------
__global__ void k_hist(const float* __restrict__ logits, uint32_t* __restrict__ hist) {
  __shared__ uint32_t h[NBIN];
  const int tid = threadIdx.x;
  for (int b = tid; b < NBIN; b += blockDim.x) h[b] = 0u;
  __syncthreads();

  const int combo = blockIdx.y;
  const int img = combo / NLVL, lvl = combo % NLVL;
  const size_t base = ((size_t)img * ATOT + d_lvl_off[lvl]) * NCLS;
  const int n4 = d_lvl_size[lvl] * NCLS / 4;   // all level sizes * 80 divisible by 4
  const float4* p = (const float4*)(logits + base);
  const int stride = gridDim.x * blockDim.x;

  for (int i = blockIdx.x * blockDim.x + tid; i < n4; i += stride) {
    __builtin_prefetch((const void*)(p + i + stride), 0, 0);  // global_prefetch_b8
    float4 v = p[i];
    if (v.x > TH_LOGIT) atomicAdd(&h[fkey(v.x) >> 21], 1u);
    if (v.y > TH_LOGIT) atomicAdd(&h[fkey(v.y) >> 21], 1u);
    if (v.z > TH_LOGIT) atomicAdd(&h[fkey(v.z) >> 21], 1u);
    if (v.w > TH_LOGIT) atomicAdd(&h[fkey(v.w) >> 21], 1u);
  }
  __syncthreads();
  uint32_t* g = hist + (size_t)combo * NBIN;
  for (int b = tid; b < NBIN; b += blockDim.x)
    if (h[b]) atomicAdd(&g[b], h[b]);
}

// ---------------- kernel 2: cutoff bin via WMMA suffix-sum ----------------
// 2048 bins -> 256 chunk sums (8 bins each). View the 256 chunk sums as a
// 16x16 matrix P; within-row inclusive suffix sums are S1 = P x L with
// L[k][n] = (k >= n), computed with 4 accumulating v_wmma_f32_16x16x4_f32.
// Counts <= 7.2M < 2^24, so f32 arithmetic is exact.
// meta layout per combo (8 uints): [0]=cutoff_bin [1]=total_passing [2]=cand_count
__global__ void k_cutoff(const uint32_t* __restrict__ hist, uint32_t* __restrict__ meta) {
  __shared__ uint32_t bins[NBIN];
  __shared__ float part[256];   // chunk sums (exact integers in f32)
  __shared__ float sfx [256];   // S1: within-row inclusive suffix sums
  const int combo = blockIdx.x;
  const int tid = threadIdx.x;
  const uint32_t* g = hist + (size_t)combo * NBIN;
  for (int b = tid; b < NBIN; b += 256) bins[b] = g[b];
  __syncthreads();
  {
    uint32_t s = 0;
    for (int k = 0; k < 8; ++k) s += bins[tid * 8 + k];
    part[tid] = (float)s;
  }
  __syncthreads();

#if __has_builtin(__builtin_amdgcn_wmma_f32_16x16x4_f32)
  if (tid < 32) {                       // wave 0, EXEC all-ones (WMMA requirement)
    const int nm = tid & 15;            // N (cols) for B/D, M (rows) for A
    const int hi = tid >> 4;            // lane half
    v8f C = {};
#pragma unroll
    for (int t = 0; t < 4; ++t) {
      // A = P[:, 4t..4t+3] (16x4): VGPR v, lane-half h -> K = 4t + 2h + v
      v2f A, B;
      A.x = part[nm * 16 + (t * 4 + hi * 2 + 0)];
      A.y = part[nm * 16 + (t * 4 + hi * 2 + 1)];
      // B = L[4t..4t+3, :] (4x16): lower-triangular ones, B[k][n] = (k >= n)
      B.x = ((t * 4 + hi * 2 + 0) >= nm) ? 1.0f : 0.0f;
      B.y = ((t * 4 + hi * 2 + 1) >= nm) ? 1.0f : 0.0f;
      C = __builtin_amdgcn_wmma_f32_16x16x4_f32(false, A, false, B,
                                                (short)0, C, false, false);
    }
    // D layout: VGPR d, lane-half h -> element S1[d + 8h][nm]
#pragma unroll
    for (int d = 0; d < 8; ++d)
      sfx[(d + hi * 8) * 16 + nm] = C[d];
  }
#else
  if (tid == 0) {
    for (int r = 0; r < 16; ++r) {
      float acc = 0.f;
      for (int j = 15; j >= 0; --j) { acc += part[r * 16 + j]; sfx[r * 16 + j] = acc; }
    }
  }
#endif
  __syncthreads();

  if (tid == 0) {
    // cross-row carry: LR[r] = sum of row totals of rows > r; rowtotal(r)=sfx[r*16]
    float LR[16]; float acc = 0.f;
    for (int r = 15; r >= 0; --r) { LR[r] = acc; acc += sfx[r * 16]; }
    uint32_t total = (uint32_t)(acc + 0.5f);
    uint32_t Keff = total < TOPK ? total : TOPK;
    uint32_t cut = 0;
    if (Keff > 0) {
      // largest chunk c whose inclusive suffix >= Keff (exists: suffix(0)=total>=Keff)
      int c = 0;
      for (c = 255; c >= 0; --c)
        if (sfx[c] + LR[c >> 4] >= (float)Keff) break;
      uint32_t cum = (c < 255) ? (uint32_t)(sfx[c + 1] + LR[(c + 1) >> 4] + 0.5f) : 0u;
      int b;
      for (b = c * 8 + 7; b >= c * 8; --b) {
        cum += bins[b];
        if (cum >= Keff) break;
      }
      cut = (uint32_t)b;
    }
    uint32_t* m = meta + combo * 8;
    m[0] = cut;
    m[1] = total;
  }
}

// ---------------- kernel 3: compact candidates >= cutoff bin ----------------
__global__ void k_compact(const float* __restrict__ logits, uint32_t* __restrict__ meta,
                          uint2* __restrict__ cand) {
  const int combo = blockIdx.y;
  const int img = combo / NLVL, lvl = combo % NLVL;
  const uint32_t cut = meta[combo * 8 + 0];
  uint32_t* cnt = &meta[combo * 8 + 2];
  uint2* cb = cand + (size_t)combo * CAP;
  const size_t base = ((size_t)img * ATOT + d_lvl_off[lvl]) * NCLS;
  const int n4 = d_lvl_size[lvl] * NCLS / 4;
  const float4* p = (const float4*)(logits + base);
  const int stride = gridDim.x * blockDim.x;
  for (int i = blockIdx.x * blockDim.x + threadIdx.x; i < n4; i += stride) {
    float4 v = p[i];
    float  xs[4] = {v.x, v.y, v.z, v.w};
#pragma unroll
    for (int c = 0; c < 4; ++c) {
      float x = xs[c];
      if (x > TH_LOGIT) {
        uint32_t k = fkey(x);
        if ((k >> 21) >= cut) {
          uint32_t pos = atomicAdd(cnt, 1u);
          if (pos < CAP) cb[pos] = make_uint2(k, (uint32_t)(i * 4 + c));
        }
      }
    }
  }
}

// ---------------- kernel 4: exact top-1000 select + decode into per-image table ----
// sel entry (32B): {score, x1, y1, x2, y2, label, area_scratch, pad}
__global__ void k_select(const float* __restrict__ regr, const float* __restrict__ anchors,
                         const uint32_t* __restrict__ meta, const uint2* __restrict__ cand,
                         float* __restrict__ sel) {
  __shared__ uint32_t skey[CAP];
  __shared__ int red[256];
  __shared__ int s_fill, s_slot;
  const int combo = blockIdx.x;
  const int tid = threadIdx.x;
  const int img = combo / NLVL, lvl = combo % NLVL;
  int N = (int)meta[combo * 8 + 2];
  if (N > CAP) N = CAP;
  const int K = N < TOPK ? N : TOPK;
  float* selL = sel + ((size_t)img * NCAND + (size_t)lvl * TOPK) * 8;

  // init all 1000 slots with filler (score = -1, matching the reference padding)
  for (int s = tid; s < TOPK; s += 256) {
    float* o = selL + (size_t)s * 8;
    o[0] = -1.0f; o[1] = 0.f; o[2] = 0.f; o[3] = 0.f;
    o[4] = 0.f;  o[5] = 0.f; o[6] = 0.f; o[7] = 0.f;
  }
  const uint2* cb = cand + (size_t)combo * CAP;
  // stage keys into LDS via the async copy engine (gather of .x fields)
  {
    const uint32_t lbase = (uint32_t)(uintptr_t)&skey[0];
    for (int i = tid; i < N; i += 256)
      async_g2l_b32(lbase + (uint32_t)i * 4u, (const void*)&cb[i].x);
    wait_async0();
  }
  if (tid == 0) { s_fill = 0; s_slot = 0; }
  __syncthreads();
  if (N == 0) return;

  // exact K-th largest key: max T with count(key >= T) >= K (bitwise descend)
  uint32_t T = 0;
  for (int bit = 31; bit >= 0; --bit) {
    uint32_t t2 = T | (1u << bit);
    int c = 0;
    for (int i = tid; i < N; i += 256) c += (skey[i] >= t2);
    red[tid] = c; __syncthreads();
    for (int off = 128; off > 0; off >>= 1) {
      if (tid < off) red[tid] += red[tid + off];
      __syncthreads();
    }
    int tot = red[0]; __syncthreads();
    if (tot >= K) T = t2;
  }
  int cgt = 0;
  for (int i = tid; i < N; i += 256) cgt += (skey[i] > T);
  red[tid] = cgt; __syncthreads();
  for (int off = 128; off > 0; off >>= 1) {
    if (tid < off) red[tid] += red[tid + off];
    __syncthreads();
  }
  const int need = K - red[0];
  __syncthreads();

  for (int i = tid; i < N; i += 256) {
    uint32_t k = skey[i];
    bool take = false;
    if (k > T) take = true;
    else if (k == T) { int p = atomicAdd(&s_fill, 1); take = (p < need); }
    if (take) {
      int slot = atomicAdd(&s_slot, 1);
      if (slot < TOPK) {
        uint32_t e = cb[i].y;
        int al  = (int)(e / NCLS);
        int cls = (int)(e - (uint32_t)al * NCLS);
        int ga  = d_lvl_off[lvl] + al;
        float x = key2f(k);
        float score = 1.0f / (1.0f + expf(-x));
        const float* d  = regr    + ((size_t)img * ATOT + ga) * 4;
        const float* an = anchors + (size_t)ga * 4;
        float wa = an[2] - an[0], ha = an[3] - an[1];
        float cxa = an[0] + 0.5f * wa, cya = an[1] + 0.5f * ha;
        float cx = d[0] * wa + cxa, cy = d[1] * ha + cya;
        float w = expf(fminf(d[2], BOXCLAMP)) * wa;
        float h = expf(fminf(d[3], BOXCLAMP)) * ha;
        float x1 = fminf(fmaxf(cx - 0.5f * w, 0.f), IMGSZ);
        float y1 = fminf(fmaxf(cy - 0.5f * h, 0.f), IMGSZ);
        float x2 = fminf(fmaxf(cx + 0.5f * w, 0.f), IMGSZ);
        float y2 = fminf(fmaxf(cy + 0.5f * h, 0.f), IMGSZ);
        float* o = selL + (size_t)slot * 8;
        o[0] = score; o[1] = x1; o[2] = y1; o[3] = x2;
        o[4] = y2; o[5] = (float)cls; o[6] = 0.f; o[7] = 0.f;
      }
    }
  }
}

// ---------------- kernel 5: per-image NMS in LDS (TDM / async staged) ----------------
__global__ void k_nms(const float* __restrict__ sel, float* __restrict__ out) {
  extern __shared__ float sm[];             // NCAND * 8 floats = 160000 B
  __shared__ float r_s[256];
  __shared__ int   r_i[256];
  const int tid = threadIdx.x;
  const int img = blockIdx.x;
  const float* src = sel + (size_t)img * NCAND * 8;
  const float NEG_INF = -__builtin_inff();

#if __has_builtin(__builtin_amdgcn_tensor_load_to_lds)
  // One TDM descriptor moves the whole 160,000 B table: 2D tensor 40000x1
  // dwords, tile == tensor, data_size = 4B. Issued by wave 0 only.
  if (tid < 32) {
    const uint64_t ga  = (uint64_t)(uintptr_t)src;
    const uint32_t lds = (uint32_t)(uintptr_t)sm;
    v4u g0 = {0u, 0u, 0u, 0u};
    g0[0] = 1u;                                   // count=1, user descriptor
    g0[1] = lds;                                  // lds_addr
    g0[2] = (uint32_t)ga;                         // global_addr[31:0]
    g0[3] = (uint32_t)(ga >> 32) | (2u << 30);    // global_addr[56:32] | type=2
    v8i g1 = {0, 0, 0, 0, 0, 0, 0, 0};
    g1[0] = (int)(2u << 16);                      // data_size = 4 bytes
    g1[1] = (int)(40000u << 16);                  // tensor_dim0[15:0] @ bits[63:48]
    g1[2] = (int)(1u << 16);                      // tensor_dim1 = 1  @ bits[111:80]
    g1[3] = (int)(40000u << 16);                  // tile_dim0 = 40000 @ bits[127:112]
    g1[4] = 1;                                    // tile_dim1 = 1
    g1[5] = 40000;                                // tensor_dim0_stride
    v4i g2 = {0, 0, 0, 0};
    v4i g3 = {0, 0, 0, 0};
#if __clang_major__ >= 23
    v8i gx = {0, 0, 0, 0, 0, 0, 0, 0};
    __builtin_amdgcn_tensor_load_to_lds(g0, g1, g2, g3, gx, 0);
#else
    __builtin_amdgcn_tensor_load_to_lds(g0, g1, g2, g3, 0);
#endif
    wait_tensor0();
  }
#else
  {
    const uint32_t lbase = (uint32_t)(uintptr_t)sm;
    for (int e = tid; e < NCAND; e += blockDim.x) {
      const char* g = (const char*)(src + (size_t)e * 8);
      uint32_t l = lbase + (uint32_t)e * 32u;
      async_g2l_b128(l,      g);
      async_g2l_b128(l + 16, g + 16);
    }
    wait_async0();
  }
#endif
  __syncthreads();

  // precompute areas into slot 6 (translation-invariant)
  for (int i = tid; i < NCAND; i += blockDim.x) {
    float* c = sm + (size_t)i * 8;
    c[6] = (c[3] - c[1]) * (c[4] - c[2]);
  }
  __syncthreads();

  for (int it = 0; it < NDET; ++it) {
    // block argmax with first-occurrence (lowest index) tie-break, like jnp.argmax
    int bi = tid;
    float bs = sm[(size_t)tid * 8];
    for (int i = tid + 256; i < NCAND; i += 256) {
      float s = sm[(size_t)i * 8];
      if (s > bs) { bs = s; bi = i; }
    }
    r_s[tid] = bs; r_i[tid] = bi;
    __syncthreads();
    for (int off = 128; off > 0; off >>= 1) {
      if (tid < off) {
        float s2 = r_s[tid + off]; int i2 = r_i[tid + off];
        if (s2 > r_s[tid] || (s2 == r_s[tid] && i2 < r_i[tid])) {
          r_s[tid] = s2; r_i[tid] = i2;
        }
      }
      __syncthreads();
    }
    const int j = r_i[0];
    const float sj = r_s[0];
    const float* cj = sm + (size_t)j * 8;
    const float jx1 = cj[1], jy1 = cj[2], jx2 = cj[3], jy2 = cj[4];
    const float jlab = cj[5], jarea = cj[6];

    if (tid == 0) {
      float* o = out + ((size_t)img * NDET + it) * 5;
      const bool valid = sj > 0.0f;
      o[0] = valid ? jx1 : 0.f;
      o[1] = valid ? jy1 : 0.f;
      o[2] = valid ? jx2 : 0.f;
      o[3] = valid ? jy2 : 0.f;
      o[4] = valid ? sj  : 0.f;
    }

    // suppress: same-class IoU > 0.5 (equivalent to the reference's
    // label*(IMG+1) coordinate-offset trick, since boxes are clipped to [0,800])
    for (int i = tid; i < NCAND; i += 256) {
      float* c = sm + (size_t)i * 8;
      float s = c[0];
      if (s != NEG_INF) {
        if (i == j) {
          c[0] = NEG_INF;
        } else if (c[5] == jlab) {
          float xx1 = fmaxf(jx1, c[1]);
          float yy1 = fmaxf(jy1, c[2]);
          float xx2 = fminf(jx2, c[3]);
          float yy2 = fminf(jy2, c[4]);
          float inter = fmaxf(xx2 - xx1, 0.f) * fmaxf(yy2 - yy1, 0.f);
          float iou = inter / (jarea + c[6] - inter + 1e-9f);
          if (iou > NMS_TH) c[0] = NEG_INF;
        }
      }
    }
    __syncthreads();
  }
}

// ---------------- host launcher ----------------
extern "C" void kernel_launch(void* const* d_in, const int* in_sizes, int n_in,
                              void* d_out, int out_size, void* d_ws, size_t ws_size,
                              hipStream_t stream) {
  (void)in_sizes; (void)n_in; (void)out_size; (void)ws_size;
  const float* logits  = (const float*)d_in[0];  // (8, 120087, 80)
  const float* regr    = (const float*)d_in[1];  // (8, 120087, 4)
  const float* anchors = (const float*)d_in[2];  // (120087, 4)
  float* out = (float*)d_out;                    // (8, 300, 5)

  // workspace layout (~4.3 MB total)
  uint32_t* hist = (uint32_t*)d_ws;                     // 40*2048 u32
  uint32_t* meta = hist + (size_t)NCOMBO * NBIN;        // 40*8 u32
  uint2*    cand = (uint2*)(meta + (size_t)NCOMBO * 8); // 40*8192 uint2 (8B aligned)
  float*    sel  = (float*)(cand + (size_t)NCOMBO * CAP); // 8*5000*8 f32 (32B aligned)

  const int nzero = NCOMBO * NBIN + NCOMBO * 8;
  k_zero<<<dim3((nzero + 255) / 256), 256, 0, stream>>>(hist, nzero);
  k_hist<<<dim3(128, NCOMBO), 256, 0, stream>>>(logits, hist);
  k_cutoff<<<NCOMBO, 256, 0, stream>>>(hist, meta);
  k_compact<<<dim3(128, NCOMBO), 256, 0, stream>>>(logits, meta, cand);
  k_select<<<NCOMBO, 256, 0, stream>>>(regr, anchors, meta, cand, sel);
  k_nms<<<NIMG, 256, NCAND * 8 * sizeof(float), stream>>>(sel, out);
}